// Router_12051678232616
// MI455X (gfx1250) — compile-verified
//
#include <hip/hip_runtime.h>
#include <math.h>

// ---------------------------------------------------------------------------
// MoE router for MI455X (gfx1250, wave32).
//   x   : [4,4096,2048] f32  -> N=16384 tokens, C=2048
//   w_g : [64,2048]     f32  -> E=64 experts
// Outputs (flat, f32): final_mask [N,2,64] | probs [N,2] | topk_idx [N,2]
//                      | final_rank [N,2] | exp_capacity (1024)
// ---------------------------------------------------------------------------

typedef float v2f __attribute__((ext_vector_type(2)));
typedef float v8f __attribute__((ext_vector_type(8)));

#define N_TOK   16384
#define C_DIM   2048
#define N_EXPT  64
#define TOPK    2
#define CAPCTY  1024
#define NCHUNK  128          // 128 chunks * 256 entries == 2*N_TOK
#define CHUNK   256

// ---------------------------------------------------------------------------
// Kernel 1: gating GEMM via V_WMMA_F32_16X16X4_F32.
// Block = 128 threads (4 waves). Block computes a 16-token x 64-expert strip;
// wave w owns experts [16w, 16w+16). K-loop: 512 WMMAs of K=4.
//
// f32 16x16x4 operand layout (ISA 7.12.2):
//   A (16x4):  lanes 0-15 -> M=lane, VGPR{0,1}=K{0,1}; lanes 16-31 -> K{2,3}
//   B (4x16):  mirrored with N=lane&15  -> identical per-lane float2 fetch
//   C/D (16x16 f32): VGPR g: lanes 0-15 -> M=g, N=lane; lanes 16-31 -> M=g+8
// ---------------------------------------------------------------------------
__global__ __launch_bounds__(128)
void router_gemm_wmma(const float* __restrict__ x,
                      const float* __restrict__ wg,
                      float* __restrict__ logits) {
  const int lane  = threadIdx.x & 31;
  const int wave  = threadIdx.x >> 5;
  const int tok0  = blockIdx.x * 16;
  const int e0    = wave * 16;
  const int r     = lane & 15;          // token row (A) / expert col (B)
  const int khalf = (lane >> 4) * 2;    // K sub-offset: 0 or 2

  const float* aptr = x  + (size_t)(tok0 + r) * C_DIM + khalf;
  const float* bptr = wg + (size_t)(e0  + r) * C_DIM + khalf;

  v8f acc = {0.f, 0.f, 0.f, 0.f, 0.f, 0.f, 0.f, 0.f};

#pragma unroll 4
  for (int kc = 0; kc < C_DIM; kc += 4) {
    v2f a = *(const v2f*)(aptr + kc);
    v2f b = *(const v2f*)(bptr + kc);
    // 8 args: (neg_a, A, neg_b, B, c_mod, C, reuse_a, reuse_b)
    acc = __builtin_amdgcn_wmma_f32_16x16x4_f32(
        false, a, false, b, (short)0, acc, false, false);
  }

  const int col     = lane & 15;
  const int rowbase = (lane >> 4) * 8;
#pragma unroll
  for (int g = 0; g < 8; ++g) {
    logits[(size_t)(tok0 + rowbase + g) * N_EXPT + e0 + col] = acc[g];
  }
}

// ---------------------------------------------------------------------------
// Kernel 2: per-token top-2 + softmax. Tie-break = lowest index (matches
// jax.lax.top_k). Also emits the capacity scalar once.
// ---------------------------------------------------------------------------
__global__ __launch_bounds__(256)
void router_top2(const float* __restrict__ logits,
                 float* __restrict__ probs,
                 int*   __restrict__ idxs,
                 float* __restrict__ out_idx,
                 float* __restrict__ out_cap) {
  const int n = blockIdx.x * 256 + threadIdx.x;
  if (n == 0 && out_cap != nullptr) *out_cap = (float)CAPCTY;
  if (n >= N_TOK) return;

  const float* row = logits + (size_t)n * N_EXPT;
  float v0 = -INFINITY; int i0 = 0;
#pragma unroll 8
  for (int e = 0; e < N_EXPT; ++e) {
    float v = row[e];
    if (v > v0) { v0 = v; i0 = e; }
  }
  float v1 = -INFINITY; int i1 = 0;
#pragma unroll 8
  for (int e = 0; e < N_EXPT; ++e) {
    if (e == i0) continue;
    float v = row[e];
    if (v > v1) { v1 = v; i1 = e; }
  }
  // softmax over (v0, v1), v0 >= v1
  float p1 = 1.0f / (1.0f + __expf(v0 - v1));
  float p0 = 1.0f - p1;

  probs[n * TOPK + 0] = p0;
  probs[n * TOPK + 1] = p1;
  idxs [n * TOPK + 0] = i0;
  idxs [n * TOPK + 1] = i1;
  out_idx[n * TOPK + 0] = (float)i0;
  out_idx[n * TOPK + 1] = (float)i1;
}

// Sequence order o in [0, 2N): k = o >= N, n = o - k*N (k-major, reference's
// transpose(1,0,2).reshape(2N, E) cumsum order).
__device__ __forceinline__ int seq_expert(const int* idxs, int o) {
  const int k = (o >= N_TOK) ? 1 : 0;
  const int n = o - k * N_TOK;
  return idxs[n * TOPK + k];
}

// ---------------------------------------------------------------------------
// Kernel 3: per-chunk expert histogram (LDS atomics).
// ---------------------------------------------------------------------------
__global__ __launch_bounds__(CHUNK)
void router_hist(const int* __restrict__ idxs, int* __restrict__ counts) {
  __shared__ int bins[N_EXPT];
  const int t = threadIdx.x;
  if (t < N_EXPT) bins[t] = 0;
  __syncthreads();
  const int o = blockIdx.x * CHUNK + t;
  atomicAdd(&bins[seq_expert(idxs, o)], 1);
  __syncthreads();
  if (t < N_EXPT) counts[blockIdx.x * N_EXPT + t] = bins[t];
}

// ---------------------------------------------------------------------------
// Kernel 4: exclusive prefix over chunks, per expert (1 block, 64 lanes).
// ---------------------------------------------------------------------------
__global__ __launch_bounds__(64)
void router_scan(int* __restrict__ counts) {
  const int e = threadIdx.x;
  if (e >= N_EXPT) return;
  int run = 0;
  for (int c = 0; c < NCHUNK; ++c) {
    const int v = counts[c * N_EXPT + e];
    counts[c * N_EXPT + e] = run;
    run += v;
  }
}

// ---------------------------------------------------------------------------
// Kernel 5: finalize. rank = chunk prefix + ordered intra-chunk count, then
// capacity mask -> one-hot mask row, masked prob, rank.
// ---------------------------------------------------------------------------
__global__ __launch_bounds__(CHUNK)
void router_finalize(const int*   __restrict__ idxs,
                     const int*   __restrict__ counts,
                     const float* __restrict__ probs,
                     float* __restrict__ out_mask,
                     float* __restrict__ out_prob,
                     float* __restrict__ out_rank) {
  __shared__ int sidx[CHUNK];
  const int t = threadIdx.x;
  const int o = blockIdx.x * CHUNK + t;
  const int k = (o >= N_TOK) ? 1 : 0;
  const int n = o - k * N_TOK;
  const int e = idxs[n * TOPK + k];
  sidx[t] = e;
  __syncthreads();

  int rank = counts[blockIdx.x * N_EXPT + e];
  for (int t2 = 0; t2 < t; ++t2) rank += (sidx[t2] == e) ? 1 : 0;

  const int  keep = (rank < CAPCTY) ? 1 : 0;
  const int  slot = n * TOPK + k;
  out_rank[slot] = (float)rank;                 // rank regardless of capacity
  out_prob[slot] = probs[slot] * (float)keep;

  float* mrow = out_mask + (size_t)slot * N_EXPT;
#pragma unroll 8
  for (int e2 = 0; e2 < N_EXPT; ++e2) mrow[e2] = 0.0f;
  if (keep) mrow[e] = 1.0f;
}

// ---------------------------------------------------------------------------
extern "C" void kernel_launch(void* const* d_in, const int* in_sizes, int n_in,
                              void* d_out, int out_size, void* d_ws, size_t ws_size,
                              hipStream_t stream) {
  const float* x  = (const float*)d_in[0];   // [16384, 2048]
  const float* wg = (const float*)d_in[1];   // [64, 2048]
  float* out = (float*)d_out;

  // workspace layout (bytes)
  float* logits = (float*)d_ws;                               // 4,194,304
  float* probs  = (float*)((char*)d_ws + 4194304);            //   131,072
  int*   idxs   = (int*)  ((char*)d_ws + 4325376);            //   131,072
  int*   counts = (int*)  ((char*)d_ws + 4456448);            //    32,768

  // output layout (f32 elements)
  float* out_mask = out;                      // 16384*2*64 = 2,097,152
  float* out_prob = out + 2097152;            // 32,768
  float* out_idx  = out + 2129920;            // 32,768
  float* out_rank = out + 2162688;            // 32,768
  float* out_cap  = (out_size > 2195456) ? (out + 2195456) : nullptr;

  router_gemm_wmma<<<N_TOK / 16, 128, 0, stream>>>(x, wg, logits);
  router_top2<<<N_TOK / 256, 256, 0, stream>>>(logits, probs, idxs, out_idx, out_cap);
  router_hist<<<NCHUNK, CHUNK, 0, stream>>>(idxs, counts);
  router_scan<<<1, 64, 0, stream>>>(counts);
  router_finalize<<<NCHUNK, CHUNK, 0, stream>>>(idxs, counts, probs,
                                                out_mask, out_prob, out_rank);
}